// GNNModel_49014166782254
// MI455X (gfx1250) — compile-verified
//
#include <hip/hip_runtime.h>
#include <math.h>

typedef __attribute__((ext_vector_type(2))) float v2f;
typedef __attribute__((ext_vector_type(8))) float v8f;

// Fast sigmoid: v_exp_f32 + v_add + v_rcp_f32 (avoids the IEEE-div expansion).
__device__ __forceinline__ float fast_sigmoid(float v) {
    return __builtin_amdgcn_rcpf(1.0f + __expf(-v));
}

// B element for the fused 16x12 * 12x6 layer-1 GEMM.
// Rows (K): 0..5 = m_in features -> W1;  6..9 = edge features -> We1; 10..11 pad.
// Cols (n): 0..2 = message (W1), 3..5 = gate logit (We1), 6..15 unused.
__device__ __forceinline__ float b_elem(int K, int n,
                                        const float* __restrict__ W1,
                                        const float* __restrict__ We1) {
    if (K < 6)  return (n < 3) ? W1[K * 3 + n] : 0.0f;
    if (K < 10) return (n >= 3 && n < 6) ? We1[(K - 6) * 3 + (n - 3)] : 0.0f;
    return 0.0f;
}

__global__ void zero_kernel(float* __restrict__ h, float* __restrict__ out,
                            int n4, int n) {
    int i = blockIdx.x * blockDim.x + threadIdx.x;
    if (i < n4) h[i] = 0.0f;
    if (i < n)  out[i] = 0.0f;
}

__global__ void relu_kernel(float* __restrict__ p, int n) {
    int i = blockIdx.x * blockDim.x + threadIdx.x;
    if (i < n) p[i] = fmaxf(p[i], 0.0f);
}

// ---------------------------------------------------------------------------
// Layer 1: one wave32 processes 16 edges via 3 chained V_WMMA_F32_16X16X4_F32.
//   A (16x4 per chunk): M = edge-in-block, K = input feature.
//     lanes 0..15 : M = lane,    VGPR0 = K0, VGPR1 = K1
//     lanes 16..31: M = lane-16, VGPR0 = K2, VGPR1 = K3
//   K layout: chunk0 = [x_src0, x_src1 | x_dst0, x_dst1]
//             chunk1 = [elev_s, elev_d | len,   geom  ]
//             chunk2 = [inoff,  outoff | 0,     0     ]
//   Accumulator seeded with per-column bias (b1 for cols 0-2, be1 for 3-5).
//   D: vgpr k, lanes 0..15 -> edge k, col = lane; lanes 16..31 -> edge k+8.
// h is stored with row stride 4 (16B) so layer 2 can gather it with B128.
// Grid must consist of fully-populated waves (EXEC all-ones for WMMA).
// ---------------------------------------------------------------------------
__global__ void __launch_bounds__(256)
edge_layer1_wmma(const int* __restrict__ srcI, const int* __restrict__ dstI,
                 const float* __restrict__ x, const float* __restrict__ elev,
                 const float* __restrict__ fLen, const float* __restrict__ fGeom,
                 const float* __restrict__ fIn,  const float* __restrict__ fOut,
                 const float* __restrict__ W1,   const float* __restrict__ b1,
                 const float* __restrict__ We1,  const float* __restrict__ be1,
                 float* __restrict__ h) {
    const int tid  = blockIdx.x * blockDim.x + threadIdx.x;
    const int lane = threadIdx.x & 31;
    const int wave = tid >> 5;
    const int e0   = wave << 4;      // 16 edges per wave
    const int r    = lane & 15;      // edge slot / B column
    const int half = lane >> 4;
    const int er   = e0 + r;

    const int s = srcI[er];
    const int d = dstI[er];

    // Broadcast dst of edge slot k to every lane early; these DS ops only
    // depend on the index load and overlap the A/B marshalling + WMMAs.
    int dstm[8];
#pragma unroll
    for (int k = 0; k < 8; ++k)
        dstm[k] = __builtin_amdgcn_ds_bpermute((k + (half << 3)) << 2, d);

    // ----- A matrix chunks -----
    const int g = half ? d : s;
    const float2 xv = ((const float2*)x)[g];
    v2f a0; a0.x = xv.x; a0.y = xv.y;
    v2f a1, a2;
    if (half) {
        a1.x = fLen[er];  a1.y = fGeom[er];
        a2.x = 0.0f;      a2.y = 0.0f;
    } else {
        a1.x = elev[s];   a1.y = elev[d];
        a2.x = fIn[er];   a2.y = fOut[er];
    }

    // ----- B matrix chunks (per-lane constants): vgpr j -> row K = 4c+2*half+j
    v2f b0, b1v, b2;
    b0.x  = b_elem(0 + 2 * half, r, W1, We1);
    b0.y  = b_elem(1 + 2 * half, r, W1, We1);
    b1v.x = b_elem(4 + 2 * half, r, W1, We1);
    b1v.y = b_elem(5 + 2 * half, r, W1, We1);
    b2.x  = b_elem(8 + 2 * half, r, W1, We1);
    b2.y  = b_elem(9 + 2 * half, r, W1, We1);

    // ----- accumulator = per-column bias, broadcast over all 16 rows -----
    const float bias = (r < 3) ? b1[r] : ((r < 6) ? be1[r - 3] : 0.0f);
    v8f acc;
#pragma unroll
    for (int i = 0; i < 8; ++i) acc[i] = bias;

    acc = __builtin_amdgcn_wmma_f32_16x16x4_f32(false, a0, false, b0,
                                                (short)0, acc, false, false);
    acc = __builtin_amdgcn_wmma_f32_16x16x4_f32(false, a1, false, b1v,
                                                (short)0, acc, false, false);
    acc = __builtin_amdgcn_wmma_f32_16x16x4_f32(false, a2, false, b2,
                                                (short)0, acc, false, false);

    // ----- sigmoid everywhere (cols 3..5 hold the gate logits) -----
    float sg[8];
#pragma unroll
    for (int k = 0; k < 8; ++k) sg[k] = fast_sigmoid(acc[k]);

    // ----- pull gate sigmoid from col n+3 into col n, multiply, scatter -----
    float val[8];
#pragma unroll
    for (int k = 0; k < 8; ++k) {
        const int gi = __builtin_amdgcn_ds_bpermute((lane + 3) << 2,
                                                    __float_as_int(sg[k]));
        val[k] = acc[k] * __int_as_float(gi);
    }
#pragma unroll
    for (int k = 0; k < 8; ++k) {
        if (r < 3) atomicAdd(&h[dstm[k] * 4 + r], val[k]);   // stride-4 rows
    }
}

// Scalar tail for layer 1 (edges not covered by full-wave WMMA blocks).
__global__ void edge_layer1_tail(const int* __restrict__ srcI,
                                 const int* __restrict__ dstI,
                                 const float* __restrict__ x,
                                 const float* __restrict__ elev,
                                 const float* __restrict__ fLen,
                                 const float* __restrict__ fGeom,
                                 const float* __restrict__ fIn,
                                 const float* __restrict__ fOut,
                                 const float* __restrict__ W1,
                                 const float* __restrict__ b1,
                                 const float* __restrict__ We1,
                                 const float* __restrict__ be1,
                                 float* __restrict__ h, int eBase, int E) {
    int e = eBase + blockIdx.x * blockDim.x + threadIdx.x;
    if (e >= E) return;
    const int s = srcI[e], d = dstI[e];
    float m[6] = { x[s * 2], x[s * 2 + 1], x[d * 2], x[d * 2 + 1],
                   elev[s], elev[d] };
    float f[4] = { fLen[e], fGeom[e], fIn[e], fOut[e] };
#pragma unroll
    for (int c = 0; c < 3; ++c) {
        float msg = b1[c];
#pragma unroll
        for (int k = 0; k < 6; ++k) msg += m[k] * W1[k * 3 + c];
        float gl = be1[c];
#pragma unroll
        for (int k = 0; k < 4; ++k) gl += f[k] * We1[k * 3 + c];
        atomicAdd(&h[d * 4 + c], msg * fast_sigmoid(gl));
    }
}

// ---------------------------------------------------------------------------
// Layer 2: out_dims = 1 -> WMMA would run at 16/256 utilization; scalar VALU
// (8 FMAs msg + 4 FMAs gate per edge) is strictly better. One thread per edge.
// h rows are 16B-aligned float4 -> single B128 gather per endpoint.
// ---------------------------------------------------------------------------
__global__ void __launch_bounds__(256)
edge_layer2(const int* __restrict__ srcI, const int* __restrict__ dstI,
            const float* __restrict__ h, const float* __restrict__ elev,
            const float* __restrict__ fLen, const float* __restrict__ fGeom,
            const float* __restrict__ fIn,  const float* __restrict__ fOut,
            const float* __restrict__ W2,   const float* __restrict__ b2,
            const float* __restrict__ We2,  const float* __restrict__ be2,
            float* __restrict__ out, int E) {
    int e = blockIdx.x * blockDim.x + threadIdx.x;
    if (e >= E) return;

    // stream-ahead hint for the hot edge arrays (global_prefetch_b8)
    if (e + 4096 < E) {
        __builtin_prefetch(&fLen[e + 4096], 0, 1);
        __builtin_prefetch(&srcI[e + 4096], 0, 1);
    }

    const int s = srcI[e], d = dstI[e];
    const float4 hs = ((const float4*)h)[s];
    const float4 hd = ((const float4*)h)[d];
    const float msg = hs.x * W2[0] + hs.y * W2[1] + hs.z * W2[2] +
                      hd.x * W2[3] + hd.y * W2[4] + hd.z * W2[5] +
                      elev[s] * W2[6] + elev[d] * W2[7] + b2[0];
    const float gl = fLen[e] * We2[0] + fGeom[e] * We2[1] +
                     fIn[e] * We2[2] + fOut[e] * We2[3] + be2[0];
    atomicAdd(&out[d], msg * fast_sigmoid(gl));
}

extern "C" void kernel_launch(void* const* d_in, const int* in_sizes, int n_in,
                              void* d_out, int out_size, void* d_ws, size_t ws_size,
                              hipStream_t stream) {
    const int*   edge_index = (const int*)d_in[0];
    const float* x    = (const float*)d_in[1];
    const float* elev = (const float*)d_in[2];
    const float* fLen = (const float*)d_in[3];
    const float* fGeom= (const float*)d_in[4];
    const float* fIn  = (const float*)d_in[5];
    const float* fOut = (const float*)d_in[6];
    const float* W1   = (const float*)d_in[7];
    const float* b1   = (const float*)d_in[8];
    const float* We1  = (const float*)d_in[9];
    const float* be1  = (const float*)d_in[10];
    const float* W2   = (const float*)d_in[11];
    const float* b2   = (const float*)d_in[12];
    const float* We2  = (const float*)d_in[13];
    const float* be2  = (const float*)d_in[14];

    const int N = in_sizes[2];        // norm_elev has N elements
    const int E = in_sizes[3];        // norm_length has E elements
    const int* srcI = edge_index;
    const int* dstI = edge_index + E;

    float* h   = (float*)d_ws;        // [N,4] padded intermediate (ch 3 = pad)
    float* out = (float*)d_out;       // [N,1]
    const int n4 = N * 4;

    zero_kernel<<<(n4 + 255) / 256, 256, 0, stream>>>(h, out, n4, N);

    // Full-wave WMMA blocks: 256 threads = 8 waves = 128 edges per block.
    const int blocks1 = E / 128;
    const int eWmma   = blocks1 * 128;
    if (blocks1 > 0) {
        edge_layer1_wmma<<<blocks1, 256, 0, stream>>>(
            srcI, dstI, x, elev, fLen, fGeom, fIn, fOut,
            W1, b1, We1, be1, h);
    }
    if (eWmma < E) {
        const int rem = E - eWmma;
        edge_layer1_tail<<<(rem + 255) / 256, 256, 0, stream>>>(
            srcI, dstI, x, elev, fLen, fGeom, fIn, fOut,
            W1, b1, We1, be1, h, eWmma, E);
    }

    relu_kernel<<<(n4 + 255) / 256, 256, 0, stream>>>(h, n4);

    edge_layer2<<<(E + 255) / 256, 256, 0, stream>>>(
        srcI, dstI, h, elev, fLen, fGeom, fIn, fOut,
        W2, b2, We2, be2, out, E);

    relu_kernel<<<(N + 255) / 256, 256, 0, stream>>>(out, N);
}